// CLUB_9302899163339
// MI455X (gfx1250) — compile-verified
//
#include <hip/hip_runtime.h>
#include <math.h>

typedef __attribute__((ext_vector_type(2))) float v2f;
typedef __attribute__((ext_vector_type(8))) float v8f;

#define NSAMP 1024
#define XDIM  512
#define YDIM  128
#define HID   1024
#define BN_EPS 1e-5f

// ---------------------------------------------------------------------------
// Column stats over `rows` + BN affine fold: a = g*rsqrt(var+eps), d = b - mean*a
// One thread per column; coalesced row-major reads; double accumulation.
// ---------------------------------------------------------------------------
__global__ void colstats_affine(const float* __restrict__ X, int rows, int cols,
                                const float* __restrict__ g, const float* __restrict__ b,
                                float* __restrict__ a, float* __restrict__ d) {
  int c = blockIdx.x * blockDim.x + threadIdx.x;
  if (c >= cols) return;
  double s = 0.0, s2 = 0.0;
  for (int r = 0; r < rows; ++r) {
    float v = X[(size_t)r * cols + c];
    s += (double)v;
    s2 += (double)v * (double)v;
  }
  float mean = (float)(s / rows);
  float var  = (float)(s2 / rows) - mean * mean;
  float av = g[c] * rsqrtf(var + BN_EPS);
  a[c] = av;
  d[c] = b[c] - mean * av;
}

// ---------------------------------------------------------------------------
// Per-column moments of y: E[y], E[y^2]  (YDIM columns, one block)
// ---------------------------------------------------------------------------
__global__ void ystats_kernel(const float* __restrict__ Y,
                              float* __restrict__ ey, float* __restrict__ ey2) {
  int c = threadIdx.x;  // launched with YDIM threads
  double s = 0.0, s2 = 0.0;
  for (int r = 0; r < NSAMP; ++r) {
    float v = Y[(size_t)r * YDIM + c];
    s += (double)v;
    s2 += (double)v * (double)v;
  }
  ey[c]  = (float)(s  / NSAMP);
  ey2[c] = (float)(s2 / NSAMP);
}

// ---------------------------------------------------------------------------
// Fold BN shift through weights: cp[j] = c0[j] + sum_k d[k] * W[k*Ncols + j]
// d staged in LDS; coalesced weight-column reads.
// ---------------------------------------------------------------------------
__global__ void biasfold(const float* __restrict__ W, const float* __restrict__ c0,
                         const float* __restrict__ dvec, int K, int Ncols,
                         float* __restrict__ cp) {
  __shared__ float sd[1024];
  for (int i = threadIdx.x; i < K; i += blockDim.x) sd[i] = dvec[i];
  __syncthreads();
  int j = blockIdx.x * blockDim.x + threadIdx.x;
  if (j >= Ncols) return;
  float acc = c0[j];
  for (int k = 0; k < K; ++k) acc += sd[k] * W[(size_t)k * Ncols + j];
  cp[j] = acc;
}

// ---------------------------------------------------------------------------
// FP32 WMMA GEMM with BN-scale folded into the A-fragment:
//   C[M,N] = act( (A * colscale) @ B + bias )   == act( A @ diag(scale)B + bias )
// One wave -> 32x64 output tile via V_WMMA_F32_16X16X4_F32:
//   2 A-fragments (row halves) x 4 shared B-fragments -> 8 independent WMMAs/step.
// A frag (16x4): lane = M row (lane&15); VGPR0/1 hold K = 2*kh + {0,1}, kh=lane>>4.
// B frag (4x16): VGPR0/1 lanes[0:16) = rows K{0,1}, lanes[16:32) = rows K{2,3}.
// C/D (16x16): VGPR r: lanes[0:16) -> M=m0+r, N=n0+lane; lanes[16:32) -> M=m0+8+r.
// Scale staged in LDS; all K-step loads batched -> one wait per step.
// EXEC stays all-ones (uniform control flow only).
// ---------------------------------------------------------------------------
__global__ void __launch_bounds__(256)
gemm_bn_wmma(const float* __restrict__ A, const float* __restrict__ B,
             const float* __restrict__ scale, const float* __restrict__ bias,
             float* __restrict__ C, int M, int K, int Ncols, int doRelu) {
  __shared__ float ssc[1024];
  for (int i = threadIdx.x; i < K; i += 256) ssc[i] = scale[i];
  __syncthreads();

  const int lane = threadIdx.x & 31;
  const int wave = threadIdx.x >> 5;
  const int colTiles = Ncols >> 6;                 // 64-wide column tiles
  const int tile = blockIdx.x * 8 + wave;          // 8 waves per block
  const int rowTile = tile / colTiles;
  const int colTile = tile % colTiles;
  const int m0 = rowTile << 5;                     // 32-row tile
  const int n0 = colTile << 6;
  const int mr = lane & 15;
  const int kh = lane >> 4;

  v8f acc[2][4] = {};

  const float* __restrict__ ap0 = A + (size_t)(m0 + mr) * K + 2 * kh;
  const float* __restrict__ ap1 = ap0 + (size_t)16 * K;
  const float* __restrict__ bp  = B + (size_t)(2 * kh) * Ncols + n0 + mr;
  const float* __restrict__ sp  = ssc + 2 * kh;
  const size_t bstep = (size_t)4 * Ncols;

  for (int k = 0; k < K; k += 4) {
    // ---- batched loads for this K-step (single wait) ----
    v2f a0, a1;
    a0.x = ap0[0]; a0.y = ap0[1];
    a1.x = ap1[0]; a1.y = ap1[1];
    float br0[4], br1[4];
#pragma unroll
    for (int t = 0; t < 4; ++t) br0[t] = bp[t * 16];
#pragma unroll
    for (int t = 0; t < 4; ++t) br1[t] = bp[Ncols + t * 16];
    const float s0 = sp[0];
    const float s1 = sp[1];

    // ---- fold BN column-scale into A (2 muls per fragment) ----
    a0.x *= s0; a0.y *= s1;
    a1.x *= s0; a1.y *= s1;

    // ---- 8 independent WMMAs (no D->A/B hazards) ----
#pragma unroll
    for (int t = 0; t < 4; ++t) {
      v2f b;
      b.x = br0[t];
      b.y = br1[t];
      acc[0][t] = __builtin_amdgcn_wmma_f32_16x16x4_f32(
          false, a0, false, b, (short)0, acc[0][t], false, false);
      acc[1][t] = __builtin_amdgcn_wmma_f32_16x16x4_f32(
          false, a1, false, b, (short)0, acc[1][t], false, false);
    }

    ap0 += 4; ap1 += 4; bp += bstep; sp += 4;
  }

  // ---- epilogue: bias + optional relu, coalesced b32 stores ----
#pragma unroll
  for (int h = 0; h < 2; ++h) {
#pragma unroll
    for (int t = 0; t < 4; ++t) {
      const int col = n0 + t * 16 + mr;
      const float bs = bias[col];
#pragma unroll
      for (int r = 0; r < 8; ++r) {
        const int row = m0 + 16 * h + 8 * kh + r;
        float v = acc[h][t][r] + bs;
        if (doRelu) v = fmaxf(v, 0.0f);
        C[(size_t)row * Ncols + col] = v;
      }
    }
  }
}

// ---------------------------------------------------------------------------
// CLUB terms + block-level tree reduction (deterministic).
// pair_mse collapsed to per-column y moments: E[y^2] - 2*mu*E[y] + mu^2.
// ---------------------------------------------------------------------------
__global__ void club_reduce(const float* __restrict__ MU, const float* __restrict__ LVP,
                            const float* __restrict__ Y, const float* __restrict__ ey,
                            const float* __restrict__ ey2, float* __restrict__ partials) {
  __shared__ float sm[256];
  const int gid = blockIdx.x * 256 + threadIdx.x;   // gid in [0, NSAMP*YDIM)
  const int k = gid & (YDIM - 1);

  const float mu = MU[gid];
  const float lv = tanhf(LVP[gid]);
  const float y  = Y[gid];
  const float diff = mu - y;
  const float pos = -diff * diff * 0.5f * expf(-2.0f * lv);
  const float pm  = ey2[k] - 2.0f * mu * ey[k] + mu * mu;
  const float neg = -pm * 0.5f * expf(-lv);

  sm[threadIdx.x] = (pos - neg) * (1.0f / (float)NSAMP);
  __syncthreads();
#pragma unroll
  for (int s = 128; s > 0; s >>= 1) {
    if (threadIdx.x < s) sm[threadIdx.x] += sm[threadIdx.x + s];
    __syncthreads();
  }
  if (threadIdx.x == 0) partials[blockIdx.x] = sm[0];
}

__global__ void final_reduce(const float* __restrict__ partials, float* __restrict__ out) {
  __shared__ float sm[512];
  sm[threadIdx.x] = partials[threadIdx.x];
  __syncthreads();
#pragma unroll
  for (int s = 256; s > 0; s >>= 1) {
    if (threadIdx.x < s) sm[threadIdx.x] += sm[threadIdx.x + s];
    __syncthreads();
  }
  if (threadIdx.x == 0) out[0] = sm[0];
}

// ---------------------------------------------------------------------------
extern "C" void kernel_launch(void* const* d_in, const int* in_sizes, int n_in,
                              void* d_out, int out_size, void* d_ws, size_t ws_size,
                              hipStream_t stream) {
  const float* x      = (const float*)d_in[0];   // [1024,512]
  const float* y      = (const float*)d_in[1];   // [1024,128]
  const float* mu_g1  = (const float*)d_in[2];
  const float* mu_b1  = (const float*)d_in[3];
  const float* mu_W1  = (const float*)d_in[4];   // [512,1024]
  const float* mu_c1  = (const float*)d_in[5];
  const float* mu_g2  = (const float*)d_in[6];
  const float* mu_b2  = (const float*)d_in[7];
  const float* mu_W2  = (const float*)d_in[8];   // [1024,128]
  const float* mu_c2  = (const float*)d_in[9];
  const float* lv_g1  = (const float*)d_in[10];
  const float* lv_b1  = (const float*)d_in[11];
  const float* lv_W1  = (const float*)d_in[12];
  const float* lv_c1  = (const float*)d_in[13];
  const float* lv_g2  = (const float*)d_in[14];
  const float* lv_b2  = (const float*)d_in[15];
  const float* lv_W2  = (const float*)d_in[16];
  const float* lv_c2  = (const float*)d_in[17];
  float* out = (float*)d_out;

  // workspace layout (floats)
  float* ws    = (float*)d_ws;
  float* a1mu  = ws;                 // 512
  float* d1mu  = a1mu + 512;
  float* a1lv  = d1mu + 512;
  float* d1lv  = a1lv + 512;
  float* ey    = d1lv + 512;         // 128
  float* ey2   = ey + 128;
  float* c1pmu = ey2 + 128;          // 1024
  float* c1plv = c1pmu + 1024;
  float* a2mu  = c1plv + 1024;       // 1024 x4
  float* d2mu  = a2mu + 1024;
  float* a2lv  = d2mu + 1024;
  float* d2lv  = a2lv + 1024;
  float* c2pmu = d2lv + 1024;        // 128 x2
  float* c2plv = c2pmu + 128;
  float* parts = c2plv + 128;        // 512
  float* Hmu   = parts + 512;        // 1024*1024
  float* Hlv   = Hmu + (size_t)NSAMP * HID;
  float* MUo   = Hlv + (size_t)NSAMP * HID;   // 1024*128
  float* LVPo  = MUo + (size_t)NSAMP * YDIM;

  // 1) BN1 stats + affine fold (both heads), y moments
  colstats_affine<<<2, 256, 0, stream>>>(x, NSAMP, XDIM, mu_g1, mu_b1, a1mu, d1mu);
  colstats_affine<<<2, 256, 0, stream>>>(x, NSAMP, XDIM, lv_g1, lv_b1, a1lv, d1lv);
  ystats_kernel<<<1, YDIM, 0, stream>>>(y, ey, ey2);

  // 2) fold BN1 shift into layer-1 bias
  biasfold<<<4, 256, 0, stream>>>(mu_W1, mu_c1, d1mu, XDIM, HID, c1pmu);
  biasfold<<<4, 256, 0, stream>>>(lv_W1, lv_c1, d1lv, XDIM, HID, c1plv);

  // 3) layer 1: H = relu(x @ diag(a1)W1 + c1')   (1024x512x1024)
  //    tiles = (1024/32)*(1024/64) = 512 waves -> 64 blocks of 8 waves
  gemm_bn_wmma<<<64, 256, 0, stream>>>(x, mu_W1, a1mu, c1pmu, Hmu, NSAMP, XDIM, HID, 1);
  gemm_bn_wmma<<<64, 256, 0, stream>>>(x, lv_W1, a1lv, c1plv, Hlv, NSAMP, XDIM, HID, 1);

  // 4) BN2 stats + affine fold
  colstats_affine<<<4, 256, 0, stream>>>(Hmu, NSAMP, HID, mu_g2, mu_b2, a2mu, d2mu);
  colstats_affine<<<4, 256, 0, stream>>>(Hlv, NSAMP, HID, lv_g2, lv_b2, a2lv, d2lv);

  // 5) fold BN2 shift into layer-2 bias
  biasfold<<<1, 256, 0, stream>>>(mu_W2, mu_c2, d2mu, HID, YDIM, c2pmu);
  biasfold<<<1, 256, 0, stream>>>(lv_W2, lv_c2, d2lv, HID, YDIM, c2plv);

  // 6) layer 2: out = H @ diag(a2)W2 + c2'   (1024x1024x128)
  //    tiles = (1024/32)*(128/64) = 64 waves -> 8 blocks
  gemm_bn_wmma<<<8, 256, 0, stream>>>(Hmu, mu_W2, a2mu, c2pmu, MUo, NSAMP, HID, YDIM, 0);
  gemm_bn_wmma<<<8, 256, 0, stream>>>(Hlv, lv_W2, a2lv, c2plv, LVPo, NSAMP, HID, YDIM, 0);

  // 7) CLUB scalar: two-stage deterministic reduction (1024*128 = 512 blocks * 256)
  club_reduce<<<512, 256, 0, stream>>>(MUo, LVPo, y, ey, ey2, parts);
  final_reduce<<<1, 512, 0, stream>>>(parts, out);
}